// Dynamic_FC_55954833932881
// MI455X (gfx1250) — compile-verified
//
#include <hip/hip_runtime.h>

// ---------------------------------------------------------------------------
// Dynamic FC with tensor-product spline basis, MI455X (gfx1250, wave32).
//   out[b,o] = relu( sum_d kb[b,d] * ( sum_i xf[b,i]*W[i,o,d] + bias[o,d] ) )
// Exact-f32 matrix path via V_WMMA_F32_16X16X4_F32; kb folded into the A
// operand; weight pre-transposed to Wt[d][i][o]; tiles staged to LDS via the
// gfx1250 async global->LDS path (ASYNCcnt) when the builtins are available.
// ---------------------------------------------------------------------------

typedef float v2f __attribute__((ext_vector_type(2)));
typedef float v8f __attribute__((ext_vector_type(8)));
typedef int   v2i __attribute__((ext_vector_type(2)));
typedef int   v4i __attribute__((ext_vector_type(4)));

#define B_ROWS 4096
#define IND    512
#define OUTD   512
#define PDIM   2
#define DBAS   5
#define DD     25
#define XCOLS  514

#define XPITCH 36    // xfs row pitch (dwords): mult of 4, 36*m mod 64 distinct
#define WPITCH 144   // wts row pitch (dwords): 144 mod 64 == 16 -> conflict-free

#if defined(__has_builtin)
#if __has_builtin(__builtin_amdgcn_global_load_async_to_lds_b64) && \
    __has_builtin(__builtin_amdgcn_global_load_async_to_lds_b128) && \
    __has_builtin(__builtin_amdgcn_s_wait_asynccnt)
#define USE_ASYNC 1
#endif
#endif

#ifdef USE_ASYNC
#define GLOBAL_AS __attribute__((address_space(1)))
#define LDS_AS    __attribute__((address_space(3)))
#define ASYNC_WAIT() __builtin_amdgcn_s_wait_asynccnt(0)
#else
#define ASYNC_WAIT()
#endif

// ---------------------------------------------------------------- prep ------
__global__ __launch_bounds__(256) void prep_kernel(const float* __restrict__ x,
                                                   float* __restrict__ kb,
                                                   float* __restrict__ out) {
  int b = blockIdx.x * 256 + threadIdx.x;
  if (b >= B_ROWS) return;
  float t0 = x[(size_t)b * XCOLS + 0];
  float t1 = x[(size_t)b * XCOLS + 1];
  float b0[DBAS], b1[DBAS], r;
  b0[0] = 1.0f; b0[1] = t0; b0[2] = t0 * t0;
  r = t0 - 0.33f; r = r > 0.0f ? r : 0.0f; b0[3] = r * r;
  r = t0 - 0.66f; r = r > 0.0f ? r : 0.0f; b0[4] = r * r;
  b1[0] = 1.0f; b1[1] = t1; b1[2] = t1 * t1;
  r = t1 - 0.33f; r = r > 0.0f ? r : 0.0f; b1[3] = r * r;
  r = t1 - 0.66f; r = r > 0.0f ? r : 0.0f; b1[4] = r * r;
#pragma unroll
  for (int i = 0; i < DBAS; ++i)
#pragma unroll
    for (int j = 0; j < DBAS; ++j)
      kb[(size_t)b * DD + i * DBAS + j] = b0[i] * b1[j];
  out[(size_t)b * XCOLS + 0] = t0;
  out[(size_t)b * XCOLS + 1] = t1;
}

// ------------------------------------------------------------ transpose -----
__global__ __launch_bounds__(256) void transpose_kernel(const float* __restrict__ w,
                                                        float* __restrict__ wt) {
  __shared__ float lds[OUTD * DD];           // 51.2 KB
  int i = blockIdx.x;
  const float* wrow = w + (size_t)i * (OUTD * DD);
  for (int e = threadIdx.x; e < OUTD * DD; e += 256) lds[e] = wrow[e];
  __syncthreads();
  for (int e = threadIdx.x; e < OUTD * DD; e += 256) {
    int o = e & (OUTD - 1);
    int d = e >> 9;
    wt[(size_t)d * (IND * OUTD) + (size_t)i * OUTD + o] = lds[o * DD + d];
  }
}

// ----------------------------------------------------------------- gemm -----
// Block tile 128(M) x 128(N); 8 waves as 2(M) x 4(N); wave tile 64x32 = 4x2
// WMMA subtiles (8 accumulators).  K in chunks of 32 over i, d inner.
__global__ __launch_bounds__(256) void gemm_kernel(const float* __restrict__ x,
                                                   const float* __restrict__ bias,
                                                   const float* __restrict__ kb,
                                                   const float* __restrict__ wt,
                                                   float* __restrict__ out) {
  __shared__ float xfs[128 * XPITCH];        // 18.4 KB
  __shared__ float wts[32 * WPITCH];         // 18.4 KB
  __shared__ float kbs[128 * DD];            // 12.8 KB

  const int t    = threadIdx.x;
  const int lane = t & 31;
  const int wid  = t >> 5;
  const int wm2  = wid & 1;                  // 2 wave rows (M)
  const int wn4  = wid >> 1;                 // 4 wave cols (N)
  const int bm0  = blockIdx.y * 128;
  const int n0   = blockIdx.x * 128;

  const int mrow = lane & 15;
  const int half = lane >> 4;

  for (int e = t; e < 128 * DD; e += 256)
    kbs[e] = kb[(size_t)(bm0 + e / DD) * DD + (e % DD)];

  v8f acc[4][2] = {};

  int am[4], bn[2];
#pragma unroll
  for (int s = 0; s < 4; ++s) am[s] = wm2 * 64 + s * 16 + mrow;
#pragma unroll
  for (int u = 0; u < 2; ++u) bn[u] = wn4 * 32 + u * 16 + mrow;

  for (int c = 0; c < IND / 32; ++c) {
    __syncthreads();                         // all prior LDS reads retired
    // ---- stage xf 128x32 (2-dword chunks; rows of x are only 8B aligned)
    {
#pragma unroll
      for (int j = 0; j < 8; ++j) {
        int ch  = t + j * 256;               // 0..2047
        int row = ch >> 4;
        int c2  = (ch & 15) * 2;
        const float* g = &x[(size_t)(bm0 + row) * XCOLS + PDIM + c * 32 + c2];
        float* l = &xfs[row * XPITCH + c2];
#ifdef USE_ASYNC
        __builtin_amdgcn_global_load_async_to_lds_b64(
            (GLOBAL_AS v2i*)g, (LDS_AS v2i*)l, 0, 0);
#else
        l[0] = g[0]; l[1] = g[1];
#endif
      }
    }
    for (int d = 0; d < DD; ++d) {
      if (d > 0) __syncthreads();            // prior wts reads retired
      // ---- stage Wt slab 32x128 (16B-aligned: b128 chunks)
      {
#pragma unroll
        for (int j = 0; j < 4; ++j) {
          int ch  = t + j * 256;             // 0..1023
          int row = ch >> 5;
          int c4  = (ch & 31) * 4;
          const float* g = &wt[(size_t)d * (IND * OUTD) +
                               (size_t)(c * 32 + row) * OUTD + n0 + c4];
          float* l = &wts[row * WPITCH + c4];
#ifdef USE_ASYNC
          __builtin_amdgcn_global_load_async_to_lds_b128(
              (GLOBAL_AS v4i*)g, (LDS_AS v4i*)l, 0, 0);
#else
          l[0] = g[0]; l[1] = g[1]; l[2] = g[2]; l[3] = g[3];
#endif
        }
      }
      ASYNC_WAIT();                          // this wave's async copies done
      __syncthreads();                       // slab visible to all waves
      float kv[4];
#pragma unroll
      for (int s = 0; s < 4; ++s) kv[s] = kbs[am[s] * DD + d];
#pragma unroll
      for (int kk = 0; kk < 8; ++kk) {
        const int kbq = kk * 4 + 2 * half;   // this lane's K pair
        v2f a[4], b[2];
#pragma unroll
        for (int s = 0; s < 4; ++s) {
          a[s].x = xfs[am[s] * XPITCH + kbq]     * kv[s];
          a[s].y = xfs[am[s] * XPITCH + kbq + 1] * kv[s];
        }
#pragma unroll
        for (int u = 0; u < 2; ++u) {
          b[u].x = wts[kbq * WPITCH + bn[u]];
          b[u].y = wts[(kbq + 1) * WPITCH + bn[u]];
        }
#pragma unroll
        for (int s = 0; s < 4; ++s)
#pragma unroll
          for (int u = 0; u < 2; ++u)
            acc[s][u] = __builtin_amdgcn_wmma_f32_16x16x4_f32(
                false, a[s], false, b[u], (short)0, acc[s][u], false, false);
      }
    }
  }

  // ---- epilogue: + sum_d kb[b,d]*bias[o,d], relu, store -------------------
  const int nc0 = n0 + wn4 * 32 + mrow;      // global o for subtile u=0
  const int nc1 = nc0 + 16;                  // subtile u=1
  float bias0[DD], bias1[DD];
#pragma unroll
  for (int d = 0; d < DD; ++d) {
    bias0[d] = bias[(size_t)nc0 * DD + d];
    bias1[d] = bias[(size_t)nc1 * DD + d];
  }
#pragma unroll
  for (int s = 0; s < 4; ++s) {
#pragma unroll
    for (int r = 0; r < 8; ++r) {
      const int m = wm2 * 64 + s * 16 + r + 8 * half;  // row within block
      float bt0 = 0.f, bt1 = 0.f;
#pragma unroll
      for (int d = 0; d < DD; ++d) {
        const float k = kbs[m * DD + d];
        bt0 += k * bias0[d];
        bt1 += k * bias1[d];
      }
      float v0 = acc[s][0][r] + bt0; v0 = v0 > 0.f ? v0 : 0.f;
      float v1 = acc[s][1][r] + bt1; v1 = v1 > 0.f ? v1 : 0.f;
      out[(size_t)(bm0 + m) * XCOLS + PDIM + nc0] = v0;
      out[(size_t)(bm0 + m) * XCOLS + PDIM + nc1] = v1;
    }
  }
}

// --------------------------------------------------------------- launch -----
extern "C" void kernel_launch(void* const* d_in, const int* in_sizes, int n_in,
                              void* d_out, int out_size, void* d_ws, size_t ws_size,
                              hipStream_t stream) {
  const float* x    = (const float*)d_in[0];   // (4096, 514)
  const float* w    = (const float*)d_in[1];   // (512, 512, 25)
  const float* bias = (const float*)d_in[2];   // (512, 25)
  float* out = (float*)d_out;                  // (4096, 514)

  // ws: kb (4096*25 f32 = 409,600 B) | Wt (25*512*512 f32 = 26.2 MB)
  float* kb = (float*)d_ws;
  float* wt = kb + (size_t)B_ROWS * DD;

  prep_kernel<<<B_ROWS / 256, 256, 0, stream>>>(x, kb, out);
  transpose_kernel<<<IND, 256, 0, stream>>>(w, wt);
  gemm_kernel<<<dim3(OUTD / 128, B_ROWS / 128), 256, 0, stream>>>(x, bias, kb, wt, out);
}